// CADenseMul_14534169330227
// MI455X (gfx1250) — compile-verified
//
#include <hip/hip_runtime.h>

typedef __attribute__((ext_vector_type(2))) float v2f;
typedef __attribute__((ext_vector_type(8))) float v8f;
typedef __attribute__((ext_vector_type(4))) int   v4i_t;

// Address-space-qualified pointer types matching the async-to-LDS builtin signatures
// (revealed by clang diagnostics: b128 wants AS1 v4i*, b32 wants AS1 int*).
typedef __attribute__((address_space(1))) v4i_t* gas_v4i;
typedef __attribute__((address_space(3))) v4i_t* las_v4i;
typedef __attribute__((address_space(1))) int*   gas_i32;
typedef __attribute__((address_space(3))) int*   las_i32;

#define B_ROWS   4096
#define D_IN     2048
#define D_CTXN   512
#define UNITS_N  2048
#define RANK_N   256

#define BLK 64        // M and N tile per workgroup
#define KT  32        // K tile
#define LDA 36        // LDS row stride (floats): 16B-aligned rows, spreads 64 banks
#define ABUF (BLK * LDA)   // one LDS buffer (floats)

// ---------------- CDNA5 async global->LDS copies (ASYNCcnt), with safe fallback ---------------
#if __has_builtin(__builtin_amdgcn_global_load_async_to_lds_b128) && \
    __has_builtin(__builtin_amdgcn_global_load_async_to_lds_b32)  && \
    __has_builtin(__builtin_amdgcn_s_wait_asynccnt)
#define HAS_ASYNC 1
#else
#define HAS_ASYNC 0
#endif

__device__ __forceinline__ void async_cp_b128(const float* g, float* l) {
#if HAS_ASYNC
  // generic->AS1 is value-preserving; LDS generic address keeps the LDS offset in [31:0].
  __builtin_amdgcn_global_load_async_to_lds_b128(
      (gas_v4i)(unsigned long long)g,
      (las_v4i)(unsigned int)(unsigned long long)l, 0, 0);
#else
  *(float4*)l = *(const float4*)g;
#endif
}

__device__ __forceinline__ void async_cp_b32(const float* g, float* l) {
#if HAS_ASYNC
  __builtin_amdgcn_global_load_async_to_lds_b32(
      (gas_i32)(unsigned long long)g,
      (las_i32)(unsigned int)(unsigned long long)l, 0, 0);
#else
  *l = *g;
#endif
}

template <int N>
__device__ __forceinline__ void async_wait() {
#if HAS_ASYNC
  __builtin_amdgcn_s_wait_asynccnt((unsigned short)N);
#endif
}

// ---------------- WMMA helpers ----------------------------------------------------------------
__device__ __forceinline__ v8f wmma4(v2f a, v2f b, v8f c) {
  return __builtin_amdgcn_wmma_f32_16x16x4_f32(false, a, false, b, (short)0, c, false, false);
}

__device__ __forceinline__ v8f vzero8() {
  v8f z;
#pragma unroll
  for (int e = 0; e < 8; ++e) z[e] = 0.0f;
  return z;
}

// Unified compute step: A tile and B^T tile both in LDS as [row][k], stride LDA.
// Every fragment is a contiguous 8-byte LDS load (no repacking movs).
__device__ __forceinline__ void mma_tile(const float* As, const float* Bt,
                                         int waveM, int waveN, int m16, int khalf,
                                         v8f acc[2][2]) {
#pragma unroll
  for (int kk = 0; kk < KT; kk += 4) {
    const int kc = kk + khalf * 2;
    v2f a0 = *(const v2f*)&As[(waveM * 32 +      m16) * LDA + kc];
    v2f a1 = *(const v2f*)&As[(waveM * 32 + 16 + m16) * LDA + kc];
    v2f b0 = *(const v2f*)&Bt[(waveN * 32 +      m16) * LDA + kc];
    v2f b1 = *(const v2f*)&Bt[(waveN * 32 + 16 + m16) * LDA + kc];
    acc[0][0] = wmma4(a0, b0, acc[0][0]);
    acc[0][1] = wmma4(a0, b1, acc[0][1]);
    acc[1][0] = wmma4(a1, b0, acc[1][0]);
    acc[1][1] = wmma4(a1, b1, acc[1][1]);
  }
}

// ---------------- Stage 1 tile issue: A [64xKT] natural, B (from [K,RANK]) transposed ---------
// 4 b128 (A) + 16 b32 (B transpose-in-flight) = 20 async ops per thread per tile.
#define NOPS_S1 20
__device__ __forceinline__ void issue_tile_s1(const float* __restrict__ Ag, int lda,
                                              const float* __restrict__ Bg,
                                              int blockM, int blockN, int k0,
                                              float* As, float* Bt, int tid) {
#pragma unroll
  for (int i = 0; i < 4; ++i) {
    const int idx = tid + i * 128;
    const int ar = idx >> 3, ac = (idx & 7) * 4;
    async_cp_b128(&Ag[(size_t)(blockM + ar) * lda + k0 + ac], &As[ar * LDA + ac]);
  }
  const int n = tid & 63;
  const int kh = (tid >> 6) * 16;
#pragma unroll
  for (int i = 0; i < 16; ++i) {
    const int k = kh + i;   // global: lanes sweep consecutive n per k-row (coalesced)
    async_cp_b32(&Bg[(size_t)(k0 + k) * RANK_N + blockN + n], &Bt[n * LDA + k]);
  }
}

// Double-buffered, async-pipelined K loop for stage 1.
__device__ __forceinline__ void gemm_s1(const float* __restrict__ Ag, int lda,
                                        const float* __restrict__ Bg, int Ktot,
                                        int blockM, int blockN,
                                        float* AsBuf, float* BtBuf,
                                        int tid, int waveM, int waveN, int m16, int khalf,
                                        v8f acc[2][2]) {
  const int T = Ktot / KT;
  issue_tile_s1(Ag, lda, Bg, blockM, blockN, 0, AsBuf, BtBuf, tid);
  for (int t = 0; t < T; ++t) {
    const int cur = t & 1;
    if (t + 1 < T) {
      issue_tile_s1(Ag, lda, Bg, blockM, blockN, (t + 1) * KT,
                    AsBuf + (1 - cur) * ABUF, BtBuf + (1 - cur) * ABUF, tid);
      async_wait<NOPS_S1>();   // retire tile t (in-order), keep tile t+1 in flight
    } else {
      async_wait<0>();
    }
    __syncthreads();           // tile t visible to all waves
    mma_tile(AsBuf + cur * ABUF, BtBuf + cur * ABUF, waveM, waveN, m16, khalf, acc);
    __syncthreads();           // all waves done reading before buffer is rewritten
  }
}

// ================= Stage 1: proj[b,r] = (x@U)[b,r] * S[r]*sigmoid((ctx@W)[b,r] + Bc[r]) =======
__global__ __launch_bounds__(128) void k_gate_proj(const float* __restrict__ x,
                                                   const float* __restrict__ ctx,
                                                   const float* __restrict__ U,
                                                   const float* __restrict__ W,
                                                   const float* __restrict__ S,
                                                   const float* __restrict__ Bc,
                                                   float* __restrict__ proj) {
  __shared__ __align__(16) float As[2 * ABUF];   // 18432 B (double-buffered)
  __shared__ __align__(16) float Bt[2 * ABUF];   // 18432 B (B^T tiles)

  const int tid   = threadIdx.x;
  const int lane  = tid & 31;
  const int wave  = tid >> 5;
  const int waveM = wave >> 1, waveN = wave & 1;
  const int m16   = lane & 15, khalf = lane >> 4;
  const int blockM = blockIdx.x * BLK;
  const int blockN = blockIdx.y * BLK;      // column offset within RANK

  v8f acc[2][2];
#pragma unroll
  for (int i = 0; i < 2; ++i)
#pragma unroll
    for (int j = 0; j < 2; ++j) acc[i][j] = vzero8();

  // ---- Phase A: h = ctx @ W (K = D_CTX) ----
  gemm_s1(ctx, D_CTXN, W, D_CTXN, blockM, blockN, As, Bt,
          tid, waveM, waveN, m16, khalf, acc);

  // ---- gate in registers: s = S[r] * sigmoid(h + Bc[r]) ----
  float sg[2][2][8];
#pragma unroll
  for (int tN = 0; tN < 2; ++tN) {
    const int r = blockN + waveN * 32 + tN * 16 + m16;
    const float Sv = S[r], Bv = Bc[r];
#pragma unroll
    for (int tM = 0; tM < 2; ++tM)
#pragma unroll
      for (int e = 0; e < 8; ++e) {
        const float h = acc[tM][tN][e] + Bv;
        sg[tM][tN][e] = Sv * (1.0f / (1.0f + __expf(-h)));
      }
  }

#pragma unroll
  for (int i = 0; i < 2; ++i)
#pragma unroll
    for (int j = 0; j < 2; ++j) acc[i][j] = vzero8();

  // ---- Phase B: p = x @ U (K = D_IN) ----
  gemm_s1(x, D_IN, U, D_IN, blockM, blockN, As, Bt,
          tid, waveM, waveN, m16, khalf, acc);

  // ---- store proj = p * s ----
#pragma unroll
  for (int tM = 0; tM < 2; ++tM)
#pragma unroll
    for (int tN = 0; tN < 2; ++tN) {
      const int col = blockN + waveN * 32 + tN * 16 + m16;
#pragma unroll
      for (int e = 0; e < 8; ++e) {
        const int row = blockM + waveM * 32 + tM * 16 + khalf * 8 + e;
        proj[(size_t)row * RANK_N + col] = acc[tM][tN][e] * sg[tM][tN][e];
      }
    }
}

// ================= Stage 2: out = proj @ V^T + bias ===========================================
// V is [UNITS, RANK] row-major => the V^T tile is naturally [n][k]: b128 async copies only.
#define NOPS_S2 8
__device__ __forceinline__ void issue_tile_s2(const float* __restrict__ proj,
                                              const float* __restrict__ V,
                                              int blockM, int blockN, int k0,
                                              float* As, float* Vs, int tid) {
#pragma unroll
  for (int i = 0; i < 4; ++i) {
    const int idx = tid + i * 128;
    const int r = idx >> 3, c = (idx & 7) * 4;
    async_cp_b128(&proj[(size_t)(blockM + r) * RANK_N + k0 + c], &As[r * LDA + c]);
    async_cp_b128(&V[(size_t)(blockN + r) * RANK_N + k0 + c], &Vs[r * LDA + c]);
  }
}

__global__ __launch_bounds__(128) void k_out(const float* __restrict__ proj,
                                             const float* __restrict__ V,
                                             const float* __restrict__ bias,
                                             float* __restrict__ out) {
  __shared__ __align__(16) float As[2 * ABUF];
  __shared__ __align__(16) float Vs[2 * ABUF];

  const int tid   = threadIdx.x;
  const int lane  = tid & 31;
  const int wave  = tid >> 5;
  const int waveM = wave >> 1, waveN = wave & 1;
  const int m16   = lane & 15, khalf = lane >> 4;
  const int blockM = blockIdx.x * BLK;
  const int blockN = blockIdx.y * BLK;      // column offset within UNITS

  v8f acc[2][2];
#pragma unroll
  for (int i = 0; i < 2; ++i)
#pragma unroll
    for (int j = 0; j < 2; ++j) acc[i][j] = vzero8();

  const int T = RANK_N / KT;   // 8
  issue_tile_s2(proj, V, blockM, blockN, 0, As, Vs, tid);
  for (int t = 0; t < T; ++t) {
    const int cur = t & 1;
    if (t + 1 < T) {
      issue_tile_s2(proj, V, blockM, blockN, (t + 1) * KT,
                    As + (1 - cur) * ABUF, Vs + (1 - cur) * ABUF, tid);
      async_wait<NOPS_S2>();
    } else {
      async_wait<0>();
    }
    __syncthreads();
    mma_tile(As + cur * ABUF, Vs + cur * ABUF, waveM, waveN, m16, khalf, acc);
    __syncthreads();
  }

#pragma unroll
  for (int tM = 0; tM < 2; ++tM)
#pragma unroll
    for (int tN = 0; tN < 2; ++tN) {
      const int col = blockN + waveN * 32 + tN * 16 + m16;
      const float bv = bias[col];
#pragma unroll
      for (int e = 0; e < 8; ++e) {
        const int row = blockM + waveM * 32 + tM * 16 + khalf * 8 + e;
        out[(size_t)row * UNITS_N + col] = acc[tM][tN][e] + bv;
      }
    }
}

extern "C" void kernel_launch(void* const* d_in, const int* in_sizes, int n_in,
                              void* d_out, int out_size, void* d_ws, size_t ws_size,
                              hipStream_t stream) {
  const float* x    = (const float*)d_in[0];  // [4096, 2048]
  const float* ctx  = (const float*)d_in[1];  // [4096, 512]
  const float* U    = (const float*)d_in[2];  // [2048, 256]
  const float* S    = (const float*)d_in[3];  // [256]
  const float* V    = (const float*)d_in[4];  // [2048, 256]
  const float* W    = (const float*)d_in[5];  // [512, 256]
  const float* Bc   = (const float*)d_in[6];  // [256]
  const float* bias = (const float*)d_in[7];  // [2048]
  float* out  = (float*)d_out;                // [4096, 2048]
  float* proj = (float*)d_ws;                 // [4096, 256] scratch (4 MB)

  dim3 g1(B_ROWS / BLK, RANK_N / BLK);        // 64 x 4
  k_gate_proj<<<g1, dim3(128), 0, stream>>>(x, ctx, U, W, S, Bc, proj);

  dim3 g2(B_ROWS / BLK, UNITS_N / BLK);       // 64 x 32
  k_out<<<g2, dim3(128), 0, stream>>>(proj, V, bias, out);
}